// InputLayer_13881334301162
// MI455X (gfx1250) — compile-verified
//
#include <hip/hip_runtime.h>
#include <hip/hip_bf16.h>

typedef __attribute__((ext_vector_type(2))) float v2f;
typedef __attribute__((ext_vector_type(8))) float v8f;

#define SIZE_D 1024
#define SCALE_F 32.0f       // sqrt(1024)
#define LN_EPS_F 1e-5f

// ---------------------------------------------------------------------------
// Kernel 1: reduce W, b (1024 elems each) and pack the WMMA B-operand.
// ws layout (floats):
//   ws[0 .. 4095]  : Bpack = 64 column-tiles x 32 lanes x float2, where for
//                    tile ct, lane j<16 holds {G1[ct*16+j], G2[ct*16+j]} and
//                    lane j>=16 holds {beta[ct*16+j-16], 0}. This is exactly
//                    the V_WMMA_F32_16X16X4_F32 B (4x16) register layout:
//                    lanes 0-15 = rows K0,K1; lanes 16-31 = rows K2,K3.
//   ws[4096] : A2 = mean(dW^2)
//   ws[4097] : C2 = mean(dW*db)
//   ws[4098] : B2 = mean(db^2)
// where dW = W - mean(W), db = b - mean(b), G1 = dW*gamma, G2 = db*gamma.
// ---------------------------------------------------------------------------
__global__ __launch_bounds__(256) void ln_stats_kernel(
    const float* __restrict__ W, const float* __restrict__ b,
    const float* __restrict__ gamma, const float* __restrict__ beta,
    float* __restrict__ ws) {
  __shared__ float red0[256];
  __shared__ float red1[256];
  __shared__ float means[2];
  const int t = threadIdx.x;

  float w[4], bv[4];
#pragma unroll
  for (int i = 0; i < 4; ++i) {
    w[i]  = W[t + 256 * i];
    bv[i] = b[t + 256 * i];
  }
  red0[t] = (w[0] + w[1]) + (w[2] + w[3]);
  red1[t] = (bv[0] + bv[1]) + (bv[2] + bv[3]);
  __syncthreads();
  for (int s = 128; s > 0; s >>= 1) {
    if (t < s) { red0[t] += red0[t + s]; red1[t] += red1[t + s]; }
    __syncthreads();
  }
  if (t == 0) {
    means[0] = red0[0] * (1.0f / 1024.0f);
    means[1] = red1[0] * (1.0f / 1024.0f);
  }
  __syncthreads();
  const float wbar = means[0], bbar = means[1];

  v2f* __restrict__ Bp = (v2f*)ws;
  float sA = 0.0f, sC = 0.0f, sB = 0.0f;
#pragma unroll
  for (int i = 0; i < 4; ++i) {
    const int idx = t + 256 * i;
    const float dw = w[i] - wbar;
    const float db = bv[i] - bbar;
    sA = fmaf(dw, dw, sA);
    sC = fmaf(dw, db, sC);
    sB = fmaf(db, db, sB);
    const float g = gamma[idx];
    const int ct = idx >> 4;
    const int j  = idx & 15;
    v2f e1; e1.x = dw * g;     e1.y = db * g;   // lanes 0-15: {G1,G2}
    v2f e2; e2.x = beta[idx];  e2.y = 0.0f;     // lanes 16-31: {beta,0}
    Bp[(ct << 5) + j]      = e1;
    Bp[(ct << 5) + 16 + j] = e2;
  }
  __syncthreads();   // done reading red0/red1 from phase 1
  red0[t] = sA; red1[t] = sC;
  __syncthreads();
  for (int s = 128; s > 0; s >>= 1) {
    if (t < s) { red0[t] += red0[t + s]; red1[t] += red1[t + s]; }
    __syncthreads();
  }
  if (t == 0) {
    ws[4096] = red0[0] * (1.0f / 1024.0f);
    ws[4097] = red1[0] * (1.0f / 1024.0f);
  }
  __syncthreads();
  red0[t] = sB;
  __syncthreads();
  for (int s = 128; s > 0; s >>= 1) {
    if (t < s) red0[t] += red0[t + s];
    __syncthreads();
  }
  if (t == 0) ws[4098] = red0[0] * (1.0f / 1024.0f);
}

// ---------------------------------------------------------------------------
// Kernel 2: each wave owns 16 rows x 256 cols (16 column tiles, fully
// unrolled). Per tile: one coalesced global_load_b64 of the pre-packed B
// operand, one V_WMMA_F32_16X16X4_F32 computing
//     out = alpha*G1 + sigma*G2 + 1*beta   (rank-3 product, exact f32),
// then 8 nontemporal b32 stores draining the D-matrix VGPRs.
// All bases are made wave-uniform via readfirstlane so addressing folds into
// immediate offsets and loop control stays scalar.
// ---------------------------------------------------------------------------
__global__ __launch_bounds__(256) void ln_expand_wmma_kernel(
    const float* __restrict__ x, const float* __restrict__ ws,
    float* __restrict__ out) {
  const int lane   = threadIdx.x & 31;
  const int lane15 = lane & 15;
  const int waveInBlk = __builtin_amdgcn_readfirstlane((int)(threadIdx.x >> 5));
  const int wave = ((int)blockIdx.x << 3) | waveInBlk;   // uniform (SGPR)
  const int rowTile  = wave >> 2;        // 4096 row tiles of 16 rows
  const int colGroup = wave & 3;         // 4 groups of 16 column tiles
  const int r0    = rowTile << 4;        // uniform
  const int ctBeg = colGroup << 4;       // uniform

  const float A2 = ws[4096];
  const float C2 = ws[4097];
  const float B2 = ws[4098];

  const float xv   = x[r0 + lane15];
  const float var  = 1024.0f * fmaf(xv, fmaf(xv, A2, 2.0f * C2), B2);
  const float rstd = rsqrtf(var + LN_EPS_F);
  const bool lo = lane < 16;
  v2f a;
  a.x = lo ? SCALE_F * xv * rstd : 1.0f;   // K0 = alpha | K2 = 1
  a.y = lo ? SCALE_F * rstd      : 0.0f;   // K1 = sigma | K3 = 0

  // B operand: pre-packed in exact register layout -> one b64 load per tile.
  const v2f* __restrict__ Bp = ((const v2f*)ws) + (ctBeg << 5) + lane;

  // Store base: uniform row base (SGPR) + small vector offset + immediates.
  float* __restrict__ outBase = out + (size_t)r0 * SIZE_D;
  const int voff = (((lane >> 4) << 3) << 10) + (ctBeg << 4) + lane15;

#pragma unroll
  for (int i = 0; i < 16; ++i) {
    const v2f bv = Bp[i << 5];             // immediate offset i*256 bytes
    v8f acc = {};
    acc = __builtin_amdgcn_wmma_f32_16x16x4_f32(
        /*neg_a=*/false, a, /*neg_b=*/false, bv,
        /*c_mod=*/(short)0, acc, /*reuse_a=*/false, /*reuse_b=*/false);
#pragma unroll
    for (int r = 0; r < 8; ++r)            // immediates: i*64 + r*4096 bytes
      __builtin_nontemporal_store(acc[r], outBase + voff + (i << 4) + (r << 10));
  }
}

// ---------------------------------------------------------------------------
// Inputs (setup_inputs order): x(2048*32), W(1024), b(1024), gamma(1024),
// beta(1024). Output: 2048*32*1024 f32 (256 MB -> ~11 us at 23.3 TB/s).
// ---------------------------------------------------------------------------
extern "C" void kernel_launch(void* const* d_in, const int* in_sizes, int n_in,
                              void* d_out, int out_size, void* d_ws, size_t ws_size,
                              hipStream_t stream) {
  (void)in_sizes; (void)n_in; (void)out_size; (void)ws_size;
  const float* x     = (const float*)d_in[0];
  const float* W     = (const float*)d_in[1];
  const float* b     = (const float*)d_in[2];
  const float* gamma = (const float*)d_in[3];
  const float* beta  = (const float*)d_in[4];
  float* out = (float*)d_out;
  float* ws  = (float*)d_ws;   // needs 4099 floats (~16.4 KB)

  ln_stats_kernel<<<1, 256, 0, stream>>>(W, b, gamma, beta, ws);
  // 65536 rows / 16 rows-per-wave * 4 col groups = 16384 waves = 2048 blocks
  ln_expand_wmma_kernel<<<2048, 256, 0, stream>>>(x, ws, out);
}